// UpPooling_70927089926351
// MI455X (gfx1250) — compile-verified
//
#include <hip/hip_runtime.h>
#include <hip/hip_bf16.h>
#include <stdint.h>
#include <math.h>

typedef float v4f __attribute__((ext_vector_type(4)));
typedef int   v4i __attribute__((ext_vector_type(4)));

#ifndef __has_builtin
#define __has_builtin(x) 0
#endif

#if __has_builtin(__builtin_amdgcn_global_load_async_to_lds_b128) && \
    __has_builtin(__builtin_amdgcn_s_wait_asynccnt)
#define USE_ASYNC_LDS 1
#else
#define USE_ASYNC_LDS 0
#endif

#define TW    512          // width (scan axis)
#define TC    64           // channels (contiguous innermost)
#define NSEG  16           // segments along W per tile
#define SEGW  (TW / NSEG)  // 32 rows per segment
#define NC4   (TC / 4)     // 16 float4 lanes across channels
#define THREADS 256

// addrspace-qualified 128-bit pointers for the async global->LDS DMA builtin
typedef __attribute__((address_space(1))) v4i* gas_v4i_ptr;
typedef __attribute__((address_space(3))) v4i* lds_v4i_ptr;

__device__ __forceinline__ v4f vmax4(v4f a, v4f b) {
    v4f r;
    r.x = fmaxf(a.x, b.x);
    r.y = fmaxf(a.y, b.y);
    r.z = fmaxf(a.z, b.z);
    r.w = fmaxf(a.w, b.w);
    return r;
}

__launch_bounds__(THREADS, 2)
__global__ void revcummax_w_kernel(const float* __restrict__ in,
                                   float* __restrict__ out) {
    extern __shared__ float tile[];          // TW*TC floats = 128 KB dynamic LDS
    __shared__ float segmax[NSEG * TC];      // 4 KB

    const int    tid  = threadIdx.x;
    const size_t base = (size_t)blockIdx.x * (size_t)(TW * TC);

    // ---------------- Phase 1: stage [W,C] slice into LDS ----------------
    // 32768 floats / (256 thr * 4) = 32 x b128 per thread, fully coalesced.
#if USE_ASYNC_LDS
    {
        gas_v4i_ptr g = (gas_v4i_ptr)(in + base);   // const cast-away: DMA read only
        lds_v4i_ptr l = (lds_v4i_ptr)tile;
        #pragma unroll
        for (int i = 0; i < (TW * TC) / (THREADS * 4); ++i) {
            const int idx = i * THREADS + tid;      // 16-byte element index
            __builtin_amdgcn_global_load_async_to_lds_b128(g + idx, l + idx,
                                                           /*offset=*/0, /*cpol=*/0);
        }
        __builtin_amdgcn_s_wait_asynccnt(0);
    }
    __syncthreads();
#else
    {
        const v4f* gin4 = (const v4f*)(in + base);
        v4f*       t4   = (v4f*)tile;
        #pragma unroll
        for (int i = 0; i < (TW * TC) / (THREADS * 4); ++i) {
            const int idx = i * THREADS + tid;
            t4[idx] = gin4[idx];
        }
    }
    __syncthreads();
#endif

    // ---------------- Phase 2: per-segment reverse scan in LDS ----------------
    const int c4 = tid & (NC4 - 1);   // 0..15 : which float4 of the channel row
    const int s  = tid >> 4;          // 0..15 : which W-segment
    v4f* t4 = (v4f*)tile;             // [TW][NC4] layout of float4

    v4f carry = { -INFINITY, -INFINITY, -INFINITY, -INFINITY };
    const int w_lo = s * SEGW;
    #pragma unroll 4
    for (int k = SEGW - 1; k >= 0; --k) {
        const int w = w_lo + k;
        v4f v = t4[w * NC4 + c4];
        carry = vmax4(carry, v);
        t4[w * NC4 + c4] = carry;     // in-place suffix max within segment
    }
    ((v4f*)segmax)[s * NC4 + c4] = carry;   // segment max (value at segment head)
    __syncthreads();

    // ---------------- Phase 3: suffix-max of later segment maxima ----------------
    v4f cin = { -INFINITY, -INFINITY, -INFINITY, -INFINITY };
    for (int sp = s + 1; sp < NSEG; ++sp) {
        cin = vmax4(cin, ((v4f*)segmax)[sp * NC4 + c4]);
    }

    // ---------------- Phase 4: apply carry, stream to global ----------------
    v4f* gout4 = (v4f*)(out + base);
    #pragma unroll 4
    for (int k = 0; k < SEGW; ++k) {
        const int w = w_lo + k;
        v4f v = vmax4(t4[w * NC4 + c4], cin);
        gout4[w * NC4 + c4] = v;
    }
}

extern "C" void kernel_launch(void* const* d_in, const int* in_sizes, int n_in,
                              void* d_out, int out_size, void* d_ws, size_t ws_size,
                              hipStream_t stream) {
    const float* in  = (const float*)d_in[0];
    float*       out = (float*)d_out;

    const int total  = in_sizes[0];          // B*H*W*C = 8*512*512*64
    const int slices = total / (TW * TC);    // B*H = 4096

    dim3 grid(slices);
    dim3 block(THREADS);
    const size_t lds_bytes = (size_t)TW * TC * sizeof(float);  // 128 KB dynamic

    hipLaunchKernelGGL(revcummax_w_kernel, grid, block, lds_bytes, stream, in, out);
}